// MorphClassifier_85710367359468
// MI455X (gfx1250) — compile-verified
//
#include <hip/hip_runtime.h>
#include <stdint.h>

// ---------------------------------------------------------------------------
// MorphClassifier: out[r] = wrap8(trunc(max(|x[r,0]+b0|, |x[r,1]+b1|)))
// Pure HBM-streaming kernel (96 MiB traffic). Data path: TDM (tensor DMA)
// HBM -> LDS, 128-bit LDS reads, 128-bit nontemporal global stores.
// ---------------------------------------------------------------------------

typedef __attribute__((ext_vector_type(4))) unsigned int u32x4;
typedef __attribute__((ext_vector_type(8))) int          i32x8;
typedef __attribute__((ext_vector_type(4))) int          i32x4;
typedef __attribute__((ext_vector_type(4))) float        f32x4;

#define ROWS_PER_BLOCK 4096   // 32 KiB of x per block
#define TPB            256    // 8 wave32s

__device__ __forceinline__ float wrap8(float m) {
  int vi = (int)m;                    // trunc toward zero == jnp.trunc -> int32
  vi = ((vi + 128) & 255) - 128;      // floored mod 256 (power of two)
  return (float)vi;
}

__global__ __launch_bounds__(TPB) void morph_classifier_kernel(
    const float* __restrict__ x, const float* __restrict__ biases,
    float* __restrict__ out, int nrows) {
  __shared__ __align__(16) float tile[ROWS_PER_BLOCK * 2];

  const int rowBase = (int)blockIdx.x * ROWS_PER_BLOCK;
  const int rem     = nrows - rowBase;
  const int n       = rem < ROWS_PER_BLOCK ? rem : ROWS_PER_BLOCK;  // rows this block
  if (n <= 0) return;
  const int nElems  = n * 2;                                        // fp32 elements
  const float b0 = biases[0];
  const float b1 = biases[1];

#if __has_builtin(__builtin_amdgcn_tensor_load_to_lds)
  // ---- Tensor Data Mover: DMA this block's slice of x into LDS (wave 0 only).
  if (threadIdx.x < 32) {
    const uint32_t lds_addr = (uint32_t)(uintptr_t)(&tile[0]);      // low 32 bits = LDS offset
    const uint64_t gaddr =
        (uint64_t)(uintptr_t)x + (uint64_t)rowBase * 2u * sizeof(float);

    // D# group 0: count=1 (valid, user mode), lds_addr, global_addr, type=2
    u32x4 g0 = {
        1u,
        lds_addr,
        (uint32_t)(gaddr & 0xffffffffu),
        (((uint32_t)(gaddr >> 32)) & 0x01ffffffu) | (2u << 30)
    };

    // D# group 1: wg_mask=0, data_size=2 (4B), 1 line of nElems elements.
    //  bits[79:48]  tensor_dim0 = nElems
    //  bits[111:80] tensor_dim1 = 1
    //  bits[127:112] tile_dim0  = nElems   (<= 8192, fits 16 bits)
    //  bits[143:128] tile_dim1  = 1
    //  bits[207:160] tensor_dim0_stride = nElems
    i32x8 g1 = {
        (int)(2u << 16),                                  // data_size = 4 bytes
        (int)(((unsigned)nElems & 0xffffu) << 16),        // tensor_dim0[15:0]
        (int)((((unsigned)nElems >> 16) & 0xffffu) | (1u << 16)), // dim0[31:16] | tensor_dim1=1
        (int)(((unsigned)nElems & 0xffffu) << 16),        // tile_dim0
        1,                                                // tile_dim1 = 1
        (int)nElems,                                      // tensor_dim0_stride[31:0]
        0, 0
    };

    i32x4 gz = {0, 0, 0, 0};
#if defined(__clang_major__) && (__clang_major__ >= 23)
    i32x8 gz8 = {0, 0, 0, 0, 0, 0, 0, 0};
    __builtin_amdgcn_tensor_load_to_lds(g0, g1, gz, gz, gz8, 0);
#else
    __builtin_amdgcn_tensor_load_to_lds(g0, g1, gz, gz, 0);
#endif
    __builtin_amdgcn_s_wait_tensorcnt(0);
  }
  __syncthreads();
#else
  // ---- Fallback: cooperative vector copy through LDS.
  {
    const f32x4* gx = (const f32x4*)(x + (size_t)rowBase * 2);
    f32x4* lt = (f32x4*)tile;
    const int nv = nElems >> 2;
    for (int i = (int)threadIdx.x; i < nv; i += TPB) lt[i] = gx[i];
    for (int e = (nv << 2) + (int)threadIdx.x; e < nElems; e += TPB)
      tile[e] = x[(size_t)rowBase * 2 + e];
  }
  __syncthreads();
#endif

  // ---- Compute: 4 rows (two LDS b128 loads) -> one b128 NT store.
  const f32x4* lt = (const f32x4*)tile;       // each f32x4 = 2 rows
  f32x4* ovec = (f32x4*)(out + (size_t)rowBase);
  const int q = n >> 2;                       // full float4 outputs this block
  for (int r4 = (int)threadIdx.x; r4 < q; r4 += TPB) {
    f32x4 a = lt[2 * r4];                     // rows 4r4, 4r4+1
    f32x4 c = lt[2 * r4 + 1];                 // rows 4r4+2, 4r4+3
    f32x4 res;
    res[0] = wrap8(fmaxf(fabsf(a[0] + b0), fabsf(a[1] + b1)));
    res[1] = wrap8(fmaxf(fabsf(a[2] + b0), fabsf(a[3] + b1)));
    res[2] = wrap8(fmaxf(fabsf(c[0] + b0), fabsf(c[1] + b1)));
    res[3] = wrap8(fmaxf(fabsf(c[2] + b0), fabsf(c[3] + b1)));
    __builtin_nontemporal_store(res, ovec + r4);
  }
  // scalar tail (none when n % 4 == 0)
  for (int r = (q << 2) + (int)threadIdx.x; r < n; r += TPB) {
    float m = fmaxf(fabsf(tile[2 * r] + b0), fabsf(tile[2 * r + 1] + b1));
    out[(size_t)rowBase + r] = wrap8(m);
  }
}

extern "C" void kernel_launch(void* const* d_in, const int* in_sizes, int n_in,
                              void* d_out, int out_size, void* d_ws, size_t ws_size,
                              hipStream_t stream) {
  (void)n_in; (void)d_ws; (void)ws_size; (void)out_size;
  const float* x      = (const float*)d_in[0];   // [B, 2] fp32
  const float* biases = (const float*)d_in[1];   // [2]    fp32
  float* out          = (float*)d_out;           // [B]    fp32

  const int rows = in_sizes[0] / 2;              // B = 8388608
  const int grid = (rows + ROWS_PER_BLOCK - 1) / ROWS_PER_BLOCK;  // 2048
  morph_classifier_kernel<<<dim3(grid), dim3(TPB), 0, stream>>>(x, biases, out, rows);
}